// QEmbedding_81681688035509
// MI455X (gfx1250) — compile-verified
//
#include <hip/hip_runtime.h>

// QEmbedding gather + fake-quant requantize for MI455X (gfx1250, wave32).
//
// out[t, :] = round(clip(weight[x[t], :] / s, -127, 127)) * s
//
// Bandwidth-bound streaming gather (~50 MB read + ~50 MB write => ~4.3 us
// floor at 23.3 TB/s). CDNA5 structure:
//   * one wave32 per token row (768 f32 = 192 x B128)
//   * row gathered with GLOBAL_LOAD_ASYNC_TO_LDS_B128 (ASYNCcnt): the whole
//     row's gather is in flight with zero VGPR load destinations, so MLP does
//     not cost register pressure -> more resident waves -> deeper HBM queue
//   * wave reads back only its own LDS region, so a single s_wait_asynccnt 0
//     is enough -- no workgroup barrier
//   * quantize in VGPRs, stream out with non-temporal B128 stores so the
//     write stream does not evict gathered weight rows from the 192 MB L2

typedef __attribute__((ext_vector_type(4))) float v4f;

#define EMB_D          768
#define QUADS_PER_ROW  (EMB_D / 4)       // 192 float4 per row
#define WAVES_PER_BLK  8                 // 256 threads
#define QUADS_PER_LANE (QUADS_PER_ROW / 32)  // 6
#define QMAX_F         127.0f

__global__ __launch_bounds__(256) void qembed_gather_kernel(
    const int* __restrict__ x,         // [n_tokens] token ids
    const float* __restrict__ weight,  // [vocab, 768]
    const float* __restrict__ scale,   // [1]
    float* __restrict__ out,           // [n_tokens, 768]
    int n_tokens)
{
    __shared__ v4f tile[WAVES_PER_BLK][QUADS_PER_ROW];   // 24 KB / block

    const int wave = threadIdx.x >> 5;
    const int lane = threadIdx.x & 31;
    const int token = blockIdx.x * WAVES_PER_BLK + wave;
    if (token >= n_tokens) return;       // wave-uniform guard

    const float s   = scale[0];          // uniform -> scalar load
    const float inv = 1.0f / s;          // one divide per thread
    const int   row = x[token];          // gather index (wave-uniform)

    const v4f* wrow = (const v4f*)(weight + (size_t)row * EMB_D);

    // Issue the whole row gather as async B128 copies into this wave's LDS
    // region. VDST = LDS byte address (low 32 bits of the generic pointer
    // per the flat-LDS aperture encoding), VADDR = 64-bit global address.
#pragma unroll
    for (int i = 0; i < QUADS_PER_LANE; ++i) {
        const int q = i * 32 + lane;
        unsigned lds_addr = (unsigned)(uintptr_t)&tile[wave][q];
        unsigned long long gaddr = (unsigned long long)(uintptr_t)(wrow + q);
        asm volatile("global_load_async_to_lds_b128 %0, %1, off"
                     :: "v"(lds_addr), "v"(gaddr)
                     : "memory");
    }

    // This wave consumes only the LDS bytes it produced: waiting on our own
    // ASYNCcnt is sufficient, no barrier needed.
    asm volatile("s_wait_asynccnt 0x0" ::: "memory");

    v4f* orow = (v4f*)(out + (size_t)token * EMB_D);

#pragma unroll
    for (int i = 0; i < QUADS_PER_LANE; ++i) {
        const int q = i * 32 + lane;
        const v4f w = tile[wave][q];     // ds_load_b128

        v4f qv;
        qv.x = __builtin_rintf(fminf(fmaxf(w.x * inv, -QMAX_F), QMAX_F)) * s;
        qv.y = __builtin_rintf(fminf(fmaxf(w.y * inv, -QMAX_F), QMAX_F)) * s;
        qv.z = __builtin_rintf(fminf(fmaxf(w.z * inv, -QMAX_F), QMAX_F)) * s;
        qv.w = __builtin_rintf(fminf(fmaxf(w.w * inv, -QMAX_F), QMAX_F)) * s;

        // Write-once stream: non-temporal store keeps weight rows in L2.
        __builtin_nontemporal_store(qv, orow + q);
    }
}

extern "C" void kernel_launch(void* const* d_in, const int* in_sizes, int n_in,
                              void* d_out, int out_size, void* d_ws, size_t ws_size,
                              hipStream_t stream) {
    const int*   x      = (const int*)d_in[0];    // token ids
    const float* weight = (const float*)d_in[1];  // [vocab, 768] fp32
    const float* scale  = (const float*)d_in[2];  // [1] fp32
    float*       out    = (float*)d_out;          // [n_tokens, 768] fp32

    const int n_tokens = in_sizes[0];             // 8 * 2048 = 16384
    const int threads  = 256;                     // 8 wave32s per block
    const int blocks   = (n_tokens + WAVES_PER_BLK - 1) / WAVES_PER_BLK;

    qembed_gather_kernel<<<blocks, threads, 0, stream>>>(x, weight, scale, out,
                                                         n_tokens);
}